// ReceiverChain_23871428231192
// MI455X (gfx1250) — compile-verified
//
#include <hip/hip_runtime.h>
#include <hip/hip_bf16.h>
#include <math.h>

// ---------------------------------------------------------------------------
// ReceiverChain fused pipeline for MI455X (gfx1250, wave32, WMMA).
//
// Pipeline:
//   1) prep_kernel     : win = sigmoid(logits); scalars a*est_w, 1-a, decay
//   2) g_kernel        : G[m] = (1/N) sum_{t<512} win[t] e^{2pi i m t / N}
//   3) m2_kernel       : M2[k][j] = sum_n coef(n,j) * G[(n-k) mod N]
//                        packed directly into real GEMM weight W (64 x 8192)
//   4) frontend_kernel : pilots -> LS -> MLP -> combined -> X (4096 x 64)
//   5) gemm_kernel     : out = X @ W via V_WMMA_F32_16X16X4_F32 (full f32),
//                        64x64 output per wave (16 accumulators, 256 WMMA),
//                        non-temporal output stores.
// ---------------------------------------------------------------------------

#define NFFT 4096
#define BATCH 4096
#define P 32
#define LCP 512
#define HID 128

typedef __attribute__((ext_vector_type(2))) float v2f;
typedef __attribute__((ext_vector_type(8))) float v8f;

// ---- stage 1: scalars + window -------------------------------------------
__global__ void prep_kernel(const float* __restrict__ logits,
                            const float* __restrict__ alpha,
                            const float* __restrict__ estw,
                            const float* __restrict__ decayp,
                            float* __restrict__ ws) {
  int t = threadIdx.x;                       // 512 threads
  ws[4 + t] = 1.0f / (1.0f + expf(-logits[t]));   // sigmoid window
  if (t == 0) {
    float a = fminf(fmaxf(alpha[0], 0.0f), 1.0f);
    ws[0] = a * estw[0];                     // a * est_w
    ws[1] = 1.0f - a;                        // (1 - a)
    ws[2] = log1pf(expf(decayp[0]));         // softplus(decay_param)
  }
}

// ---- stage 2: circulant generator G --------------------------------------
__global__ void g_kernel(const float* __restrict__ ws,
                         float* __restrict__ G_re,
                         float* __restrict__ G_im) {
  __shared__ float win[LCP];
  for (int i = threadIdx.x; i < LCP; i += blockDim.x) win[i] = ws[4 + i];
  __syncthreads();
  int m = blockIdx.x * blockDim.x + threadIdx.x;   // 0..4095
  float sr = 0.0f, si = 0.0f;
  for (int t = 0; t < LCP; ++t) {
    int p = (m * t) & (NFFT - 1);            // exact mod-N phase reduction
    float ang = (float)p * 1.5339807878856412e-3f;  // 2*pi/4096
    float s, c;
    __sincosf(ang, &s, &c);
    sr += win[t] * c;
    si += win[t] * s;
  }
  G_re[m] = sr * (1.0f / (float)NFFT);
  G_im[m] = si * (1.0f / (float)NFFT);
}

// ---- stage 3: fused interp + circulant -> GEMM weight --------------------
// W is 64 x 8192 row-major.  Rows 2j/2j+1 = (re, im) of pilot j; columns
// 2k/2k+1 = (re, im) of output bin k, encoding complex multiplication.
__global__ void m2_kernel(const float* __restrict__ scal,
                          const float* __restrict__ G_re,
                          const float* __restrict__ G_im,
                          float* __restrict__ W) {
  int k = blockIdx.x * blockDim.x + threadIdx.x;   // 0..4095
  int j = blockIdx.y;                              // 0..31
  float decay = scal[2];
  int nlo = (j > 0) ? (j - 1) * 128 : 0;
  int nhi = (j >= 30) ? NFFT : (j + 1) * 128;
  float m_re = 0.0f, m_im = 0.0f;
  for (int n = nlo; n < nhi; ++n) {
    int L = n >> 7; if (L > 31) L = 31;
    int R = L + 1;
    float x0 = (float)(L << 7);
    float x1 = (R < 32) ? (float)(R << 7) : 4095.0f;
    float wl = expf(-decay * ((float)n - x0));
    float wr = expf(-decay * (x1 - (float)n));
    float inv = 1.0f / (wl + wr + 1e-12f);
    float coef = 0.0f;
    if (R < 32) {
      if (L == j) coef += wl * inv;
      if (R == j) coef += wr * inv;
    } else {  // n in [3968,4096): L==31, extrapolated pilot folds to 31 & 30
      if (j == 31) coef += wl * inv + wr * inv * (255.0f / 128.0f);
      else         coef -= wr * inv * (127.0f / 128.0f);  // j == 30
    }
    int idx = (n - k) & (NFFT - 1);          // coalesced: k contiguous
    m_re += coef * G_re[idx];
    m_im += coef * G_im[idx];
  }
  W[(size_t)(2 * j)     * 8192 + 2 * k    ] =  m_re;
  W[(size_t)(2 * j + 1) * 8192 + 2 * k    ] = -m_im;
  W[(size_t)(2 * j)     * 8192 + 2 * k + 1] =  m_im;
  W[(size_t)(2 * j + 1) * 8192 + 2 * k + 1] =  m_re;
}

// ---- stage 4: pilots -> LS -> MLP -> combined ----------------------------
__global__ void __launch_bounds__(128)
frontend_kernel(const float* __restrict__ Y_re, const float* __restrict__ Y_im,
                const float* __restrict__ Xp_re, const float* __restrict__ Xp_im,
                const int* __restrict__ pilot_pos,
                const float* __restrict__ W1, const float* __restrict__ b1,
                const float* __restrict__ W2, const float* __restrict__ b2,
                const float* __restrict__ W3, const float* __restrict__ b3,
                const float* __restrict__ scal, float* __restrict__ X) {
  __shared__ float ls[2 * P];
  __shared__ float h1[HID];
  __shared__ float h2[HID];
  __shared__ float o3[2 * P];
  int b = blockIdx.x;
  int t = threadIdx.x;                       // 0..127
  if (t < P) {
    int p = pilot_pos[t];
    float yr = Y_re[(size_t)b * NFFT + p];
    float yi = Y_im[(size_t)b * NFFT + p];
    float xr = Xp_re[t], xi = Xp_im[t];
    float d = xr * xr + xi * xi;
    ls[2 * t]     = (yr * xr + yi * xi) / d;   // LS real
    ls[2 * t + 1] = (yi * xr - yr * xi) / d;   // LS imag
  }
  __syncthreads();
  {
    float acc = b1[t];
    for (int i = 0; i < 2 * P; ++i) acc += ls[i] * W1[i * HID + t];
    h1[t] = fmaxf(acc, 0.0f);
  }
  __syncthreads();
  {
    float acc = b2[t];
    for (int i = 0; i < HID; ++i) acc += h1[i] * W2[i * HID + t];
    h2[t] = fmaxf(acc, 0.0f);
  }
  __syncthreads();
  if (t < 2 * P) {
    float acc = b3[t];
    for (int i = 0; i < HID; ++i) acc += h2[i] * W3[i * (2 * P) + t];
    o3[t] = tanhf(acc);
  }
  __syncthreads();
  if (t < P) {
    float aw = scal[0], oma = scal[1];
    X[(size_t)b * 64 + 2 * t]     = aw * ls[2 * t]     + oma * o3[t];
    X[(size_t)b * 64 + 2 * t + 1] = aw * ls[2 * t + 1] + oma * o3[t + P];
  }
}

// ---- stage 5: X (4096x64) @ W (64x8192) via f32 WMMA ---------------------
// One wave computes a 64x64 output block: 4x4 grid of 16x16 tiles, K=64
// swept in 16 steps of V_WMMA_F32_16X16X4_F32 (256 WMMA per wave).  This
// cuts A/B fragment traffic 4x vs one-tile-per-wave (~268 MB total from L2,
// balanced against the 134 MB non-temporal output store stream).
__global__ void __launch_bounds__(128)
gemm_kernel(const float* __restrict__ X, const float* __restrict__ W,
            float* __restrict__ out) {
  const int lane = threadIdx.x;              // 0..31 (wave32)
  const int half = lane >> 4;                // 0 | 1
  const int l16  = lane & 15;
  const int mbase = (blockIdx.y * 4 + threadIdx.y) * 64;
  const int nbase = blockIdx.x * 64;

  // A 16x4 f32 frag: lanes 0-15 hold K={0,1}, lanes 16-31 hold K={2,3}
  const float* Aptr = X + (size_t)(mbase + l16) * 64 + 2 * half;
  // B 4x16 f32 frag (mirror): lanes 0-15 K={0,1}, lanes 16-31 K={2,3}, N=l16
  const float* Bptr = W + (size_t)(2 * half) * 8192 + nbase + l16;

  v8f acc[4][4] = {};
#pragma unroll
  for (int kk = 0; kk < 16; ++kk) {
    v2f av[4], bv[4];
#pragma unroll
    for (int mi = 0; mi < 4; ++mi) {
      av[mi].x = Aptr[(size_t)mi * 16 * 64 + kk * 4 + 0];
      av[mi].y = Aptr[(size_t)mi * 16 * 64 + kk * 4 + 1];
    }
#pragma unroll
    for (int ni = 0; ni < 4; ++ni) {
      bv[ni].x = Bptr[(size_t)(kk * 4 + 0) * 8192 + ni * 16];
      bv[ni].y = Bptr[(size_t)(kk * 4 + 1) * 8192 + ni * 16];
    }
#pragma unroll
    for (int mi = 0; mi < 4; ++mi) {
#pragma unroll
      for (int ni = 0; ni < 4; ++ni) {
        acc[mi][ni] = __builtin_amdgcn_wmma_f32_16x16x4_f32(
            /*neg_a=*/false, av[mi], /*neg_b=*/false, bv[ni],
            /*c_mod=*/(short)0, acc[mi][ni],
            /*reuse_a=*/false, /*reuse_b=*/false);
      }
    }
  }

  // C/D layout: VGPR r -> row r (lanes 0-15) / row r+8 (lanes 16-31).
  // Output is write-once / never re-read on device -> non-temporal stores.
#pragma unroll
  for (int mi = 0; mi < 4; ++mi) {
#pragma unroll
    for (int r = 0; r < 8; ++r) {
      size_t row = (size_t)(mbase + mi * 16 + r + 8 * half);
#pragma unroll
      for (int ni = 0; ni < 4; ++ni) {
        __builtin_nontemporal_store(
            acc[mi][ni][r], &out[row * 8192 + nbase + ni * 16 + l16]);
      }
    }
  }
}

// ---------------------------------------------------------------------------
extern "C" void kernel_launch(void* const* d_in, const int* in_sizes, int n_in,
                              void* d_out, int out_size, void* d_ws, size_t ws_size,
                              hipStream_t stream) {
  (void)in_sizes; (void)n_in; (void)out_size; (void)ws_size;
  const float* Y_re    = (const float*)d_in[0];
  const float* Y_im    = (const float*)d_in[1];
  const float* Xp_re   = (const float*)d_in[2];
  const float* Xp_im   = (const float*)d_in[3];
  const int*   pilots  = (const int*)  d_in[4];
  const float* W1      = (const float*)d_in[5];
  const float* b1      = (const float*)d_in[6];
  const float* W2      = (const float*)d_in[7];
  const float* b2      = (const float*)d_in[8];
  const float* W3      = (const float*)d_in[9];
  const float* b3      = (const float*)d_in[10];
  const float* est_w   = (const float*)d_in[11];
  const float* alpha   = (const float*)d_in[12];
  const float* decayp  = (const float*)d_in[13];
  const float* wlogits = (const float*)d_in[14];

  // Workspace layout (floats), total ~3.1 MB:
  float* ws   = (float*)d_ws;
  float* scal = ws;                 // [0]=a*est_w [1]=1-a [2]=decay
  float* G_re = ws + 1024;          // 4096
  float* G_im = ws + 1024 + 4096;   // 4096
  float* X    = ws + 16384;                  // 4096*64
  float* Wm   = ws + 16384 + BATCH * 64;     // 64*8192
  float* out  = (float*)d_out;

  prep_kernel<<<1, LCP, 0, stream>>>(wlogits, alpha, est_w, decayp, ws);
  g_kernel<<<NFFT / 256, 256, 0, stream>>>(ws, G_re, G_im);
  m2_kernel<<<dim3(NFFT / 256, P), 256, 0, stream>>>(scal, G_re, G_im, Wm);
  frontend_kernel<<<BATCH, HID, 0, stream>>>(Y_re, Y_im, Xp_re, Xp_im, pilots,
                                             W1, b1, W2, b2, W3, b3, scal, X);
  gemm_kernel<<<dim3(128, 16), dim3(32, 4), 0, stream>>>(X, Wm, out);
}